// OnnxAttAdd_17489106829844
// MI455X (gfx1250) — compile-verified
//
#include <hip/hip_runtime.h>
#include <hip/hip_bf16.h>
#include <math.h>

typedef float v2f __attribute__((ext_vector_type(2)));
typedef float v8f __attribute__((ext_vector_type(8)));

#define T_LEN   16384
#define ADIM    1024
#define DUNITS_ 1024
#define EPROJ_  1024
#define SCALING_F 2.0f

// ---------------------------------------------------------------------------
// K1: dec_att = dec_z @ W_dec + b_dec using V_WMMA_F32_16X16X4_F32.
// One wave per 16-column tile (64 waves total). A-matrix has only M=0 valid
// (dec_z); rows 1..15 are zeroed via branch-free v_cndmask selects (loads are
// unconditional so EXEC stays all-1s around every WMMA, as the ISA requires).
// 32-bit A 16x4 layout: lanes 0-15: M=lane; VGPR0=K(2*half), VGPR1=K(2*half+1)
// with half=lane>>4. B 4x16 mirrors. D row M=0 = acc[0] on lanes 0-15.
// ---------------------------------------------------------------------------
__global__ __launch_bounds__(32) void k1_dec_gemv_wmma(
    const float* __restrict__ dec_z, const float* __restrict__ W_dec,
    const float* __restrict__ b_dec, float* __restrict__ dec_att)
{
    const int lane  = threadIdx.x & 31;
    const int m     = lane & 15;
    const int half  = lane >> 4;          // 0 -> K=0,1 ; 1 -> K=2,3
    const int n0    = blockIdx.x * 16;
    const bool row0 = (m == 0);           // loop-invariant predicate

    v8f acc = {0.f, 0.f, 0.f, 0.f, 0.f, 0.f, 0.f, 0.f};

    for (int k = 0; k < DUNITS_; k += 4) {
        const int kk = k + 2 * half;
        const float z0 = dec_z[kk + 0];   // unconditional: no EXEC branching
        const float z1 = dec_z[kk + 1];
        v2f a, b;
        a.x = row0 ? z0 : 0.f;            // v_cndmask, branch-free
        a.y = row0 ? z1 : 0.f;
        b.x = W_dec[(size_t)(kk + 0) * ADIM + n0 + m];
        b.y = W_dec[(size_t)(kk + 1) * ADIM + n0 + m];
        acc = __builtin_amdgcn_wmma_f32_16x16x4_f32(
            /*neg_a=*/false, a, /*neg_b=*/false, b,
            /*c_mod=*/(short)0, acc, /*reuse_a=*/false, /*reuse_b=*/false);
    }
    if (half == 0) dec_att[n0 + m] = acc[0] + b_dec[n0 + m];
}

// ---------------------------------------------------------------------------
// K2: e_scaled[t] = SCALING*(sum_a tanh(pre[t,a]+dec_att[a])*w_gvec[a]
//                            + b_gvec + mask[t])
// One wave per row; float4 (b128) streaming loads; wave shfl reduction.
// 2048 blocks x 8 waves keeps >2k waves in flight: bandwidth-limited stream.
// ---------------------------------------------------------------------------
__global__ __launch_bounds__(256) void k2_energy(
    const float* __restrict__ pre, const float* __restrict__ dec_att,
    const float* __restrict__ w_gvec, const float* __restrict__ b_gvec,
    const float* __restrict__ mask, float* __restrict__ e_scaled)
{
    const int lane = threadIdx.x & 31;
    const int wave = threadIdx.x >> 5;
    const int t    = blockIdx.x * 8 + wave;

    const float4* pre4 = (const float4*)(pre + (size_t)t * ADIM);
    const float4* d4   = (const float4*)dec_att;
    const float4* g4   = (const float4*)w_gvec;

    float sum = 0.f;
    #pragma unroll 2
    for (int i = lane; i < ADIM / 4; i += 32) {
        const float4 p = pre4[i];
        const float4 d = d4[i];
        const float4 g = g4[i];
        sum += tanhf(p.x + d.x) * g.x;
        sum += tanhf(p.y + d.y) * g.y;
        sum += tanhf(p.z + d.z) * g.z;
        sum += tanhf(p.w + d.w) * g.w;
    }
    #pragma unroll
    for (int off = 16; off > 0; off >>= 1)
        sum += __shfl_xor(sum, off, 32);

    if (lane == 0)
        e_scaled[t] = SCALING_F * (sum + b_gvec[0] + mask[t]);
}

// ---------------------------------------------------------------------------
// K3: softmax over 16384 energies; single block, fixed reduction order
// (deterministic). Writes normalized weights to w_out.
// ---------------------------------------------------------------------------
__global__ __launch_bounds__(1024) void k3_softmax(
    float* __restrict__ e_scaled, float* __restrict__ w_out)
{
    __shared__ float red[1024];
    const int tid = threadIdx.x;

    float m = -3.402823466e38f;
    for (int i = tid; i < T_LEN; i += 1024) m = fmaxf(m, e_scaled[i]);
    red[tid] = m; __syncthreads();
    for (int s = 512; s > 0; s >>= 1) {
        if (tid < s) red[tid] = fmaxf(red[tid], red[tid + s]);
        __syncthreads();
    }
    const float gmax = red[0];
    __syncthreads();

    float sum = 0.f;
    for (int i = tid; i < T_LEN; i += 1024) {
        const float ex = expf(e_scaled[i] - gmax);
        e_scaled[i] = ex;          // cache exp for the normalize pass
        sum += ex;
    }
    red[tid] = sum; __syncthreads();
    for (int s = 512; s > 0; s >>= 1) {
        if (tid < s) red[tid] += red[tid + s];
        __syncthreads();
    }
    const float inv = 1.f / red[0];
    __syncthreads();

    for (int i = tid; i < T_LEN; i += 1024)
        w_out[i] = e_scaled[i] * inv;
}

// ---------------------------------------------------------------------------
// K4: partial context over 256-row chunks: part[chunk, j] =
//     sum_{t in chunk} enc_h[t, j] * w[t].  Coalesced column reads; w loads
//     are uniform -> scalar path. grid = (EPROJ/256, 64).
// ---------------------------------------------------------------------------
__global__ __launch_bounds__(256) void k4_context_partial(
    const float* __restrict__ enc_h, const float* __restrict__ w,
    float* __restrict__ part)
{
    const int j  = blockIdx.x * 256 + threadIdx.x;
    const int t0 = blockIdx.y * 256;
    float acc = 0.f;
    #pragma unroll 4
    for (int tt = 0; tt < 256; ++tt)
        acc += enc_h[(size_t)(t0 + tt) * EPROJ_ + j] * w[t0 + tt];
    part[(size_t)blockIdx.y * EPROJ_ + j] = acc;
}

// ---------------------------------------------------------------------------
// K5: c[j] = sum over 64 chunks (fixed order -> deterministic).
// ---------------------------------------------------------------------------
__global__ __launch_bounds__(256) void k5_context_reduce(
    const float* __restrict__ part, float* __restrict__ c)
{
    const int j = blockIdx.x * 256 + threadIdx.x;
    float acc = 0.f;
    #pragma unroll
    for (int ch = 0; ch < 64; ++ch)
        acc += part[(size_t)ch * EPROJ_ + j];
    c[j] = acc;
}

extern "C" void kernel_launch(void* const* d_in, const int* in_sizes, int n_in,
                              void* d_out, int out_size, void* d_ws, size_t ws_size,
                              hipStream_t stream)
{
    (void)in_sizes; (void)n_in; (void)out_size; (void)ws_size;

    const float* dec_z  = (const float*)d_in[0];
    /* d_in[1] = att_prev: unused by the reference */
    const float* pre    = (const float*)d_in[2];
    const float* enc_h  = (const float*)d_in[3];
    const float* mask   = (const float*)d_in[4];
    const float* W_dec  = (const float*)d_in[5];
    const float* b_dec  = (const float*)d_in[6];
    const float* w_gvec = (const float*)d_in[7];
    const float* b_gvec = (const float*)d_in[8];

    float* out_c = (float*)d_out;            // [1024]
    float* out_w = (float*)d_out + EPROJ_;   // [16384]

    float* ws          = (float*)d_ws;
    float* ws_dec_att  = ws;                       // 1024 floats
    float* ws_e        = ws + 1024;                // 16384 floats
    float* ws_part     = ws + 1024 + T_LEN;        // 64*1024 floats

    k1_dec_gemv_wmma<<<ADIM / 16, 32, 0, stream>>>(dec_z, W_dec, b_dec, ws_dec_att);
    k2_energy<<<T_LEN / 8, 256, 0, stream>>>(pre, ws_dec_att, w_gvec, b_gvec, mask, ws_e);
    k3_softmax<<<1, 1024, 0, stream>>>(ws_e, out_w);
    k4_context_partial<<<dim3(EPROJ_ / 256, 64), 256, 0, stream>>>(enc_h, out_w, ws_part);
    k5_context_reduce<<<EPROJ_ / 256, 256, 0, stream>>>(ws_part, out_c);
}